// DHYPR_77936476553782
// MI455X (gfx1250) — compile-verified
//
#include <hip/hip_runtime.h>
#include <hip/hip_bf16.h>

#define MIN_NORM 1e-15f
#define PROJ_EPS 4e-3f

typedef __attribute__((ext_vector_type(2))) float v2f;
typedef __attribute__((ext_vector_type(8))) float v8f;

// ---------------- wave32 helpers ----------------

__device__ __forceinline__ float wsum(float v) {
#pragma unroll
  for (int o = 16; o > 0; o >>= 1) v += __shfl_xor(v, o, 32);
  return v;
}

__device__ __forceinline__ float artanh_(float x) {
  x = fminf(fmaxf(x, -1.0f + 1e-7f), 1.0f - 1e-7f);
  return atanhf(x);
}

template <int V>
__device__ __forceinline__ float rnorm(const float (&v)[V]) {
  float s = 0.f;
#pragma unroll
  for (int i = 0; i < V; ++i) s += v[i] * v[i];
  return fmaxf(sqrtf(wsum(s)), MIN_NORM);
}

template <int V>
__device__ __forceinline__ void proj_ip(float (&v)[V]) {
  float n = rnorm(v);
  const float maxn = 1.0f - PROJ_EPS;
  if (n > maxn) {
    float s = maxn / n;
#pragma unroll
    for (int i = 0; i < V; ++i) v[i] *= s;
  }
}

template <int V>
__device__ __forceinline__ void expmap0_ip(float (&v)[V]) {
  float n = rnorm(v);
  float s = tanhf(n) / n;
#pragma unroll
  for (int i = 0; i < V; ++i) v[i] *= s;
}

template <int V>
__device__ __forceinline__ void logmap0_ip(float (&v)[V]) {
  float n = rnorm(v);
  float s = artanh_(n) / n;
#pragma unroll
  for (int i = 0; i < V; ++i) v[i] *= s;
}

template <int V>
__device__ __forceinline__ void mulscaler_ip(float (&v)[V], float r) {
  float n = rnorm(v);
  float s = tanhf(r * artanh_(n)) / n;
#pragma unroll
  for (int i = 0; i < V; ++i) v[i] *= s;
}

// x <- mobius_add(x, y), c = 1
template <int V>
__device__ __forceinline__ void mobius_add_ip(float (&x)[V], const float (&y)[V]) {
  float x2 = 0.f, y2 = 0.f, xy = 0.f;
#pragma unroll
  for (int i = 0; i < V; ++i) {
    x2 += x[i] * x[i];
    y2 += y[i] * y[i];
    xy += x[i] * y[i];
  }
  x2 = wsum(x2);
  y2 = wsum(y2);
  xy = wsum(xy);
  float den = fmaxf(1.f + 2.f * xy + x2 * y2, MIN_NORM);
  float ca = 1.f + 2.f * xy + y2;
  float cb = 1.f - x2;
#pragma unroll
  for (int i = 0; i < V; ++i) x[i] = (ca * x[i] + cb * y[i]) / den;
}

// ---------------- kernels ----------------

// out = proj(expmap0(in)), row-wise, one wave (32 lanes) per row of D floats
template <int D>
__global__ __launch_bounds__(256) void k_exp_proj(const float* __restrict__ in,
                                                  float* __restrict__ outp, int N) {
  int row = (int)(((unsigned)blockIdx.x * blockDim.x + threadIdx.x) >> 5);
  int lane = threadIdx.x & 31;
  if (row >= N) return;
  constexpr int V = D / 32;
  float v[V];
  const float* r = in + (size_t)row * D;
#pragma unroll
  for (int i = 0; i < V; ++i) v[i] = r[lane + i * 32];
  expmap0_ip(v);
  proj_ip(v);
  float* o = outp + (size_t)row * D;
#pragma unroll
  for (int i = 0; i < V; ++i) o[lane + i * 32] = v[i];
}

// Y[N, OUT] = X[N, 128] @ W[OUT, 128]^T using V_WMMA_F32_16X16X4_F32.
// One wave computes a 16-row block across all OUT columns (OUT/16 tiles).
// A frag (16x4 f32): lane L<16 -> A[L][k..k+1], lane>=16 -> A[L-16][k+2..k+3].
// B frag (4x16 f32): lane L<16 -> W[L][k..k+1],  lane>=16 -> W[L-16][k+2..k+3].
template <int OUT>
__global__ __launch_bounds__(128) void k_gemm_wmma(const float* __restrict__ X,
                                                   const float* __restrict__ W,
                                                   float* __restrict__ Y, int N) {
  constexpr int IN = 128;
  constexpr int NT = OUT / 16;
  int wave = threadIdx.x >> 5;
  int lane = threadIdx.x & 31;
  int rowblk = blockIdx.x * 4 + wave;
  int row0 = rowblk * 16;
  if (row0 >= N) return;  // wave-uniform exit: EXEC stays all-ones for WMMA

  const v8f vzero = {0.f, 0.f, 0.f, 0.f, 0.f, 0.f, 0.f, 0.f};
  v8f acc[NT];
#pragma unroll
  for (int t = 0; t < NT; ++t) acc[t] = vzero;

  int m = lane & 15;
  int kh = (lane >> 4) * 2;  // 0 or 2
  int mrow = row0 + m;
  if (mrow >= N) mrow = N - 1;  // clamp (duplicate loads, guarded stores)
  const float* xrow = X + (size_t)mrow * IN;

#pragma unroll 4
  for (int k = 0; k < IN; k += 4) {
    v2f a = *(const v2f*)(xrow + k + kh);
#pragma unroll
    for (int t = 0; t < NT; ++t) {
      const float* wrow = W + (size_t)(t * 16 + m) * IN;
      v2f b = *(const v2f*)(wrow + k + kh);
      acc[t] = __builtin_amdgcn_wmma_f32_16x16x4_f32(
          false, a, false, b, (short)0, acc[t], false, false);
    }
  }

  // C/D layout: VGPR r: lanes 0-15 -> M=r, lanes 16-31 -> M=r+8, N=lane&15
  int nIdx = lane & 15;
  int mofs = (lane >> 4) * 8;
#pragma unroll
  for (int t = 0; t < NT; ++t) {
#pragma unroll
    for (int r = 0; r < 8; ++r) {
      int rr = row0 + mofs + r;
      if (rr < N) Y[(size_t)rr * OUT + t * 16 + nIdx] = acc[t][r];
    }
  }
}

// Fused: mobius_matvec tail + proj + mobius_add(bias) + proj + logmap0.
// mx[N,D] in/out (becomes tangent xt); hin[N,128] is the layer input (for xn).
template <int D>
__global__ __launch_bounds__(256) void k_linear_post(float* __restrict__ mx,
                                                     const float* __restrict__ hin,
                                                     const float* __restrict__ bias,
                                                     int N) {
  int row = (int)(((unsigned)blockIdx.x * blockDim.x + threadIdx.x) >> 5);
  int lane = threadIdx.x & 31;
  if (row >= N) return;
  constexpr int V = D / 32;

  // xn = ||h_in row|| over 128 dims
  float s = 0.f;
  const float* hr = hin + (size_t)row * 128;
#pragma unroll
  for (int i = 0; i < 4; ++i) {
    float t = hr[lane + i * 32];
    s += t * t;
  }
  float xn = fmaxf(sqrtf(wsum(s)), MIN_NORM);

  float* mr = mx + (size_t)row * D;
  float m[V];
  float s2 = 0.f, nzc = 0.f;
#pragma unroll
  for (int i = 0; i < V; ++i) {
    m[i] = mr[lane + i * 32];
    s2 += m[i] * m[i];
    nzc += (m[i] != 0.f) ? 1.f : 0.f;
  }
  s2 = wsum(s2);
  nzc = wsum(nzc);
  float mxn = fmaxf(sqrtf(s2), MIN_NORM);
  float sc = tanhf(mxn / xn * artanh_(xn)) / mxn;
  if (nzc == 0.f) sc = 0.f;  // all-zero mx row -> zero result

  float h[V];
#pragma unroll
  for (int i = 0; i < V; ++i) h[i] = m[i] * sc;
  proj_ip(h);

  // hb = proj(expmap0(bias))
  float bv[V];
#pragma unroll
  for (int i = 0; i < V; ++i) bv[i] = bias[lane + i * 32];
  expmap0_ip(bv);
  proj_ip(bv);

  mobius_add_ip(h, bv);
  proj_ip(h);
  logmap0_ip(h);  // tangent vector for aggregation

#pragma unroll
  for (int i = 0; i < V; ++i) mr[lane + i * 32] = h[i];
}

// st[dst[e]] += w[e] * xt[src[e]]; one wave per edge, atomics on f32.
template <int D>
__global__ __launch_bounds__(256) void k_spmm(const int* __restrict__ src,
                                              const int* __restrict__ dst,
                                              const float* __restrict__ w,
                                              const float* __restrict__ xt,
                                              float* __restrict__ st, int E) {
  int e = (int)(((unsigned)blockIdx.x * blockDim.x + threadIdx.x) >> 5);
  int lane = threadIdx.x & 31;
  if (e >= E) return;
  int sI = src[e];
  int dI = dst[e];
  float we = w[e];
  const float* xr = xt + (size_t)sI * D;
  float* orow = st + (size_t)dI * D;
#pragma unroll
  for (int i = 0; i < D / 32; ++i)
    atomicAdd(&orow[lane + i * 32], we * xr[lane + i * 32]);
}

// h = proj(expmap0(st)); act: proj(expmap0(relu(logmap0(h))))
template <int D>
__global__ __launch_bounds__(256) void k_agg_act(const float* __restrict__ in,
                                                 float* __restrict__ outp, int N) {
  int row = (int)(((unsigned)blockIdx.x * blockDim.x + threadIdx.x) >> 5);
  int lane = threadIdx.x & 31;
  if (row >= N) return;
  constexpr int V = D / 32;
  float v[V];
  const float* r = in + (size_t)row * D;
#pragma unroll
  for (int i = 0; i < V; ++i) v[i] = r[lane + i * 32];
  expmap0_ip(v);
  proj_ip(v);
  logmap0_ip(v);
#pragma unroll
  for (int i = 0; i < V; ++i) v[i] = fmaxf(v[i], 0.f);
  expmap0_ip(v);
  proj_ip(v);
  float* o = outp + (size_t)row * D;
#pragma unroll
  for (int i = 0; i < V; ++i) o[lane + i * 32] = v[i];
}

// Branch combination + tangent mean + final expmap/proj. D = 64 (V = 2).
__global__ __launch_bounds__(256) void k_combine(const float* __restrict__ br,
                                                 float* __restrict__ outp, int N) {
  int row = (int)(((unsigned)blockIdx.x * blockDim.x + threadIdx.x) >> 5);
  int lane = threadIdx.x & 31;
  if (row >= N) return;

  float b[4][2];
#pragma unroll
  for (int j = 0; j < 4; ++j) {
    const float* r = br + (size_t)j * N * 64 + (size_t)row * 64;
    b[j][0] = r[lane];
    b[j][1] = r[lane + 32];
  }

  // target = chain of mobius_add over mobius_mulscaler(0.125, b_j)
  float tgt[2] = {b[0][0], b[0][1]};
  mulscaler_ip(tgt, 0.125f);
#pragma unroll
  for (int j = 1; j < 4; ++j) {
    float ms[2] = {b[j][0], b[j][1]};
    mulscaler_ip(ms, 0.125f);
    mobius_add_ip(tgt, ms);
  }

  // agg = mean over logmap0 of {b0..b3, target}
  float agg[2] = {0.f, 0.f};
#pragma unroll
  for (int j = 0; j < 4; ++j) {
    float t[2] = {b[j][0], b[j][1]};
    logmap0_ip(t);
    agg[0] += t[0];
    agg[1] += t[1];
  }
  {
    float t[2] = {tgt[0], tgt[1]};
    logmap0_ip(t);
    agg[0] += t[0];
    agg[1] += t[1];
  }
  agg[0] *= 0.2f;
  agg[1] *= 0.2f;

  expmap0_ip(agg);
  proj_ip(agg);
  float* o = outp + (size_t)row * 64;
  o[lane] = agg[0];
  o[lane + 32] = agg[1];
}

// ---------------- host launch ----------------

extern "C" void kernel_launch(void* const* d_in, const int* in_sizes, int n_in,
                              void* d_out, int out_size, void* d_ws, size_t ws_size,
                              hipStream_t stream) {
  const float* x = (const float*)d_in[0];
  const int* esrc = (const int*)d_in[1];
  const int* edst = (const int*)d_in[2];
  const float* ew = (const float*)d_in[3];
  const float* W0 = (const float*)d_in[4];
  const float* b0 = (const float*)d_in[5];
  const float* W1 = (const float*)d_in[6];
  const float* b1 = (const float*)d_in[7];
  float* out = (float*)d_out;

  const int FIN = 128, HID = 128, DIM = 64, NB = 4;
  const int n = in_sizes[0] / FIN;
  const int E = in_sizes[1];

  // Workspace layout (floats): h0[n*128] | t0[n*128] | t1[n*128] | br[4*n*64]
  float* h0 = (float*)d_ws;
  float* t0 = h0 + (size_t)n * FIN;
  float* t1 = t0 + (size_t)n * HID;
  float* br = t1 + (size_t)n * HID;

  const int rowB = 256;                      // 8 waves = 8 rows per block
  const int rowG = (n + 7) / 8;
  const int gemmRB = (n + 15) / 16;          // 16-row blocks
  const int gemmG = (gemmRB + 3) / 4;        // 4 waves per block
  const int spmmG = (E + 7) / 8;             // 8 edges per block

  // h0 = proj(expmap0(x))
  k_exp_proj<128><<<rowG, rowB, 0, stream>>>(x, h0, n);

  for (int i = 0; i < NB; ++i) {
    const float* Wi0 = W0 + (size_t)i * HID * FIN;
    const float* bi0 = b0 + (size_t)i * HID;
    const float* Wi1 = W1 + (size_t)i * DIM * HID;
    const float* bi1 = b1 + (size_t)i * DIM;
    const float* wi = ew + (size_t)i * E;
    float* bri = br + (size_t)i * n * DIM;

    // ---- layer 0: 128 -> 128 ----
    k_gemm_wmma<128><<<gemmG, 128, 0, stream>>>(h0, Wi0, t1, n);
    k_linear_post<128><<<rowG, rowB, 0, stream>>>(t1, h0, bi0, n);
    hipMemsetAsync(t0, 0, (size_t)n * HID * sizeof(float), stream);
    k_spmm<128><<<spmmG, 256, 0, stream>>>(esrc, edst, wi, t1, t0, E);
    k_agg_act<128><<<rowG, rowB, 0, stream>>>(t0, t0, n);

    // ---- layer 1: 128 -> 64 ----
    k_gemm_wmma<64><<<gemmG, 128, 0, stream>>>(t0, Wi1, t1, n);
    k_linear_post<64><<<rowG, rowB, 0, stream>>>(t1, t0, bi1, n);
    hipMemsetAsync(bri, 0, (size_t)n * DIM * sizeof(float), stream);
    k_spmm<64><<<spmmG, 256, 0, stream>>>(esrc, edst, wi, t1, bri, E);
    k_agg_act<64><<<rowG, rowB, 0, stream>>>(bri, bri, n);
  }

  k_combine<<<rowG, rowB, 0, stream>>>(br, out, n);

  (void)n_in;
  (void)out_size;
  (void)ws_size;
}